// PsiCritic_10892037062685
// MI455X (gfx1250) — compile-verified
//
#include <hip/hip_runtime.h>

typedef __attribute__((ext_vector_type(2))) float v2f;
typedef __attribute__((ext_vector_type(4))) float v4f;
typedef __attribute__((ext_vector_type(8))) float v8f;

#define B_DIM  4096
#define SF     256
#define FC     512
#define ADIM   16
#define TILE_B 32   // 32 batch rows per block: two 16-row WMMA tiles sharing B-fragments

// Accumulate two 16x16 f32 tiles (rows [0,16) and [16,32) of the LDS tile)
// against the same B columns [n0, n0+16). A (32 x K) from LDS row-major,
// B (K x N) from global row-major (stride ldw).
// V_WMMA_F32_16X16X4_F32, K stepped by 4; each loaded B-fragment feeds 2 WMMAs.
// A-frag layout (ISA 7.12.2, 32-bit A 16x4): lane M = lane&15;
//   lanes 0-15: VGPR0=K0, VGPR1=K1 ; lanes 16-31: VGPR0=K2, VGPR1=K3
//   -> per lane a contiguous K pair => single ds_load_b64.
// B-frag (4x16): lane N = lane&15;
//   VGPR0 = W[(k0+hi*2)*ldw + n], VGPR1 = W[(k0+hi*2+1)*ldw + n].
__device__ __forceinline__ void gemm16x2(const float* __restrict__ ldsA, int K,
                                         const float* __restrict__ W, int ldw,
                                         int n0, int lane, v8f& acc0, v8f& acc1)
{
    const int lo = lane & 15;
    const int hi = lane >> 4;                     // 0 or 1
    const float* Arow0 = ldsA + lo * K;           // rows 0..15
    const float* Arow1 = ldsA + (lo + 16) * K;    // rows 16..31
    const float* Wcol  = W + n0 + lo;             // N = n0 + lo
#pragma unroll 4
    for (int k0 = 0; k0 < K; k0 += 4) {
        const int kk = k0 + hi * 2;
        v2f a0 = *(const v2f*)(Arow0 + kk);       // ds_load_b64
        v2f a1 = *(const v2f*)(Arow1 + kk);       // ds_load_b64
        v2f b;
        b.x = Wcol[kk * ldw];                     // global_load_b32
        b.y = Wcol[(kk + 1) * ldw];               // global_load_b32
        acc0 = __builtin_amdgcn_wmma_f32_16x16x4_f32(
            false, a0, false, b, (short)0, acc0, false, false);
        acc1 = __builtin_amdgcn_wmma_f32_16x16x4_f32(
            false, a1, false, b, (short)0, acc1, false, false);
    }
}

// C/D 16x16 f32 layout: VGPR i holds row M = i + hi*8, col N = n0 + lo.
__device__ __forceinline__ void store_relu_lds2(v8f acc0, v8f acc1,
                                                float* __restrict__ ldsOut,
                                                int ldo, int n0, int lane,
                                                const float* __restrict__ bias)
{
    const int lo = lane & 15;
    const int hi = lane >> 4;
    const float bv = bias[n0 + lo];
#pragma unroll
    for (int i = 0; i < 8; ++i) {
        float v0 = fmaxf(acc0[i] + bv, 0.0f);
        float v1 = fmaxf(acc1[i] + bv, 0.0f);
        ldsOut[(i + hi * 8) * ldo + n0 + lo]        = v0;
        ldsOut[(i + hi * 8 + 16) * ldo + n0 + lo]   = v1;
    }
}

__global__ __launch_bounds__(256) void psi_mlp_kernel(
    const float* __restrict__ x,
    const float* __restrict__ W1, const float* __restrict__ b1,
    const float* __restrict__ W2, const float* __restrict__ b2,
    const float* __restrict__ W3, const float* __restrict__ b3,
    const float* __restrict__ V1, const float* __restrict__ c1,
    const float* __restrict__ V2, const float* __restrict__ c2,
    const float* __restrict__ V3, const float* __restrict__ c3,
    float* __restrict__ out, float* __restrict__ psi_state)
{
    __shared__ float lds_x [TILE_B * SF];   // 32 KB
    __shared__ float lds_h1[TILE_B * FC];   // 64 KB
    __shared__ float lds_h2[TILE_B * FC];   // 64 KB  (160 KB total: 2 WGs / WGP)

    const int tile   = blockIdx.x;          // 0..127 : batch tile (32 rows)
    const int branch = blockIdx.y;          // 0..16  : 16 actions + state MLP
    const int b0     = tile * TILE_B;
    const int tid    = threadIdx.x;
    const int lane   = tid & 31;
    const int wave   = tid >> 5;            // 0..7

    // --- stage x tile into LDS (contiguous 32*256 floats, 128-bit loads) ---
    const v4f* xsrc = (const v4f*)(x + (size_t)b0 * SF);
    v4f* xdst = (v4f*)lds_x;
#pragma unroll
    for (int i = tid; i < TILE_B * SF / 4; i += 256)
        xdst[i] = xsrc[i];
    __syncthreads();

    const bool is_state = (branch == ADIM);  // block-uniform
    const float* Wl1 = is_state ? V1 : (W1 + (size_t)branch * SF * FC);
    const float* Bl1 = is_state ? c1 : (b1 + branch * FC);
    const float* Wl2 = is_state ? V2 : (W2 + (size_t)branch * FC * FC);
    const float* Bl2 = is_state ? c2 : (b2 + branch * FC);
    const float* Wl3 = is_state ? V3 : (W3 + (size_t)branch * FC * SF);
    const float* Bl3 = is_state ? c3 : (b3 + branch * SF);

    // --- layer 1: relu([32,256] @ [256,512] + b) -> lds_h1 ---
    for (int j = 0; j < 4; ++j) {
        const int n0 = (wave * 4 + j) * 16;
        v8f acc0 = {}, acc1 = {};
        gemm16x2(lds_x, SF, Wl1, FC, n0, lane, acc0, acc1);
        store_relu_lds2(acc0, acc1, lds_h1, FC, n0, lane, Bl1);
    }
    __syncthreads();

    // --- layer 2: relu([32,512] @ [512,512] + b) -> lds_h2 ---
    for (int j = 0; j < 4; ++j) {
        const int n0 = (wave * 4 + j) * 16;
        v8f acc0 = {}, acc1 = {};
        gemm16x2(lds_h1, FC, Wl2, FC, n0, lane, acc0, acc1);
        store_relu_lds2(acc0, acc1, lds_h2, FC, n0, lane, Bl2);
    }
    __syncthreads();

    // --- layer 3: [32,512] @ [512,256] + b -> global ---
    for (int j = 0; j < 2; ++j) {
        const int n0 = (wave * 2 + j) * 16;
        v8f acc0 = {}, acc1 = {};
        gemm16x2(lds_h2, FC, Wl3, SF, n0, lane, acc0, acc1);
        const int lo = lane & 15;
        const int hi = lane >> 4;
        const float bv = Bl3[n0 + lo];
        if (is_state) {
#pragma unroll
            for (int i = 0; i < 8; ++i) {
                const int r = i + hi * 8;
                psi_state[(size_t)(b0 + r)      * SF + n0 + lo] = acc0[i] + bv;
                psi_state[(size_t)(b0 + r + 16) * SF + n0 + lo] = acc1[i] + bv;
            }
        } else {
#pragma unroll
            for (int i = 0; i < 8; ++i) {
                const int r = i + hi * 8;
                out[((size_t)(b0 + r)      * ADIM + branch) * SF + n0 + lo] = acc0[i] + bv;
                out[((size_t)(b0 + r + 16) * ADIM + branch) * SF + n0 + lo] = acc1[i] + bv;
            }
        }
    }
}

// psi = psi_state + adv - mean_a(adv), done in place on d_out.
__global__ __launch_bounds__(256) void psi_combine_kernel(
    float* __restrict__ out, const float* __restrict__ psi_state)
{
    const int idx = blockIdx.x * 256 + threadIdx.x;   // b*256 + n
    const int b = idx >> 8;
    const int n = idx & 255;
    const float ps = psi_state[idx];
    float* base = out + ((size_t)b * ADIM) * SF + n;
    float v[ADIM];
    float sum = 0.0f;
#pragma unroll
    for (int a = 0; a < ADIM; ++a) { v[a] = base[a * SF]; sum += v[a]; }
    const float mean = sum * (1.0f / ADIM);
#pragma unroll
    for (int a = 0; a < ADIM; ++a) base[a * SF] = v[a] - mean + ps;
}

extern "C" void kernel_launch(void* const* d_in, const int* in_sizes, int n_in,
                              void* d_out, int out_size, void* d_ws, size_t ws_size,
                              hipStream_t stream)
{
    const float* x  = (const float*)d_in[0];
    const float* W1 = (const float*)d_in[1];
    const float* b1 = (const float*)d_in[2];
    const float* W2 = (const float*)d_in[3];
    const float* b2 = (const float*)d_in[4];
    const float* W3 = (const float*)d_in[5];
    const float* b3 = (const float*)d_in[6];
    const float* V1 = (const float*)d_in[7];
    const float* c1 = (const float*)d_in[8];
    const float* V2 = (const float*)d_in[9];
    const float* c2 = (const float*)d_in[10];
    const float* V3 = (const float*)d_in[11];
    const float* c3 = (const float*)d_in[12];

    float* out       = (float*)d_out;
    float* psi_state = (float*)d_ws;     // B*SF floats = 4 MB scratch

    dim3 grid(B_DIM / TILE_B, ADIM + 1); // 128 batch tiles x (16 actions + state)
    psi_mlp_kernel<<<grid, 256, 0, stream>>>(x, W1, b1, W2, b2, W3, b3,
                                             V1, c1, V2, c2, V3, c3,
                                             out, psi_state);

    psi_combine_kernel<<<(B_DIM * SF) / 256, 256, 0, stream>>>(out, psi_state);
}